// NERMRCModel_15934328668963
// MI455X (gfx1250) — compile-verified
//
#include <hip/hip_runtime.h>
#include <cmath>

#define H_DIM 768
#define INTER 1024
#define NB 2
#define LSEQ 256
#define ROWS (NB * LSEQ)   // 512

typedef __attribute__((ext_vector_type(2))) float v2f;
typedef __attribute__((ext_vector_type(8))) float v8f;

// ---------------------------------------------------------------------------
// Kernel 1: start/end logits. One wave32 per (b,l) row.
// ---------------------------------------------------------------------------
__global__ void logits_kernel(const float* __restrict__ ctx,
                              const float* __restrict__ w_start,
                              const float* __restrict__ b_start,
                              const float* __restrict__ w_end,
                              const float* __restrict__ b_end,
                              float* __restrict__ out) {
  const int wave = (blockIdx.x * blockDim.x + threadIdx.x) >> 5;
  const int lane = threadIdx.x & 31;
  if (wave >= ROWS) return;
  const float* row = ctx + (size_t)wave * H_DIM;
  float s0 = 0.f, s1 = 0.f;
  for (int k = lane; k < H_DIM; k += 32) {
    const float c = row[k];
    s0 += c * w_start[k];
    s1 += c * w_end[k];
  }
  for (int off = 16; off > 0; off >>= 1) {
    s0 += __shfl_xor(s0, off, 32);
    s1 += __shfl_xor(s1, off, 32);
  }
  if (lane == 0) {
    out[wave]        = s0 + b_start[0];
    out[ROWS + wave] = s1 + b_end[0];
  }
}

// ---------------------------------------------------------------------------
// Kernel 2: projection GEMMs via V_WMMA_F32_16X16X4_F32 (fp32, matches ref).
//   ps = X @ W1[0:768]   (+ b1 folded in)   -> psb
//   pe = X @ W1[768:1536]                   -> pe
// One wave per 16x16 tile. 4096 tiles total, 8 waves/block, 512 blocks.
// A 16x4 f32 layout: lanes 0-15 = M rows, lane-half selects K pair (K=2*half).
// B 4x16 f32 layout mirrors A with N on lanes.
// C/D: VGPR v of lane l holds M = m0 + v + 8*(l>>4), N = n0 + (l&15).
// ---------------------------------------------------------------------------
__global__ void proj_wmma_kernel(const float* __restrict__ X,    // ROWS x H
                                 const float* __restrict__ W1,   // 2H x INTER
                                 const float* __restrict__ b1,   // INTER
                                 float* __restrict__ psb,        // ROWS x INTER
                                 float* __restrict__ pe) {       // ROWS x INTER
  const int lane = threadIdx.x & 31;
  const int wave = threadIdx.x >> 5;
  const int tile = blockIdx.x * 8 + wave;     // 0..4095
  const int mat  = tile >> 11;                // 0: ps, 1: pe  (2048 tiles each)
  const int rem  = tile & 2047;
  const int m0   = (rem >> 6) << 4;           // 32 row tiles
  const int n0   = (rem & 63) << 4;           // 64 col tiles
  const int half = lane >> 4;
  const int lr   = lane & 15;

  const float* Arow = X + (size_t)(m0 + lr) * H_DIM + 2 * half;
  const float* Bcol = W1 + (size_t)mat * H_DIM * INTER
                         + (size_t)(2 * half) * INTER + (n0 + lr);

  v8f c = {};
  for (int k = 0; k < H_DIM; k += 4) {
    v2f a;
    a.x = Arow[k];
    a.y = Arow[k + 1];
    v2f b;
    b.x = Bcol[(size_t)k * INTER];
    b.y = Bcol[(size_t)k * INTER + INTER];
    c = __builtin_amdgcn_wmma_f32_16x16x4_f32(
        /*neg_a=*/false, a, /*neg_b=*/false, b,
        /*c_mod=*/(short)0, c, /*reuse_a=*/false, /*reuse_b=*/false);
  }

  const float bias = (mat == 0) ? b1[n0 + lr] : 0.f;
  float* dst = (mat == 0) ? psb : pe;
#pragma unroll
  for (int v = 0; v < 8; ++v) {
    const int r = m0 + v + 8 * half;
    dst[(size_t)r * INTER + n0 + lr] = c[v] + bias;
  }
}

// ---------------------------------------------------------------------------
// Kernel 3: match logits. 16x16 (i,j) tile per 256-thread block.
//   out[b,i,j] = b2 + sum_k gelu(psb[b,i,k] + pe[b,j,k]) * W2[k]
// k staged through LDS in 64-chunks; rows padded to 65 floats (bank conflicts).
// ---------------------------------------------------------------------------
__global__ void match_kernel(const float* __restrict__ psb,
                             const float* __restrict__ pe,
                             const float* __restrict__ W2,
                             const float* __restrict__ b2,
                             float* __restrict__ out) {
  __shared__ float psS[16][65];
  __shared__ float peS[16][65];
  __shared__ float w2S[64];

  const int tid = threadIdx.x;
  const int b   = blockIdx.x >> 8;       // 256 tiles per batch
  const int rem = blockIdx.x & 255;
  const int i0  = (rem >> 4) << 4;
  const int j0  = (rem & 15) << 4;
  const int ti  = tid >> 4;
  const int tj  = tid & 15;
  const int rowbase = b * LSEQ;

  float acc = 0.f;
  for (int k0 = 0; k0 < INTER; k0 += 64) {
    __syncthreads();
    for (int f = tid; f < 16 * 64; f += 256) {
      const int r  = f >> 6;
      const int kk = f & 63;
      psS[r][kk] = psb[(size_t)(rowbase + i0 + r) * INTER + k0 + kk];
      peS[r][kk] = pe [(size_t)(rowbase + j0 + r) * INTER + k0 + kk];
    }
    if (tid < 64) w2S[tid] = W2[k0 + tid];
    __syncthreads();

#pragma unroll 4
    for (int kk = 0; kk < 64; ++kk) {
      const float x = psS[ti][kk] + peS[tj][kk];
      const float g = 0.5f * x * (1.0f + erff(x * 0.70710678118654752f));
      acc += g * w2S[kk];
    }
  }

  out[(size_t)b * (LSEQ * LSEQ) + (i0 + ti) * LSEQ + (j0 + tj)] = acc + b2[0];
}

// ---------------------------------------------------------------------------
extern "C" void kernel_launch(void* const* d_in, const int* in_sizes, int n_in,
                              void* d_out, int out_size, void* d_ws, size_t ws_size,
                              hipStream_t stream) {
  (void)in_sizes; (void)n_in; (void)out_size; (void)ws_size;
  const float* ctx     = (const float*)d_in[0];
  const float* w_start = (const float*)d_in[1];
  const float* b_start = (const float*)d_in[2];
  const float* w_end   = (const float*)d_in[3];
  const float* b_end   = (const float*)d_in[4];
  const float* W1      = (const float*)d_in[5];
  const float* b1      = (const float*)d_in[6];
  const float* W2      = (const float*)d_in[7];
  const float* b2      = (const float*)d_in[8];

  float* out = (float*)d_out;
  float* psb = (float*)d_ws;                       // ROWS x INTER (ps + b1)
  float* pe  = psb + (size_t)ROWS * INTER;         // ROWS x INTER

  // 512 waves -> 64 blocks of 256 threads
  logits_kernel<<<64, 256, 0, stream>>>(ctx, w_start, b_start, w_end, b_end, out);

  // 4096 tiles, 8 waves/block -> 512 blocks
  proj_wmma_kernel<<<512, 256, 0, stream>>>(ctx, W1, b1, psb, pe);

  // 2 * 16 * 16 = 512 tiles
  match_kernel<<<512, 256, 0, stream>>>(psb, pe, W2, b2, out + 2 * ROWS);
}